// VariableLengthOutputSelection_22505628631391
// MI455X (gfx1250) — compile-verified
//
#include <hip/hip_runtime.h>
#include <stdint.h>

// Problem constants from the reference: B=64, T=2048, H=512
#define B_ 64
#define T_ 2048
#define H_ 512

// One block per batch row. 256 threads:
//   tid [0,128)   -> 16B chunk tid of row lstm1[b, input_length[b]-1, :]
//   tid [128,256) -> 16B chunk (tid-128) of row lstm2[b, support_length[b]-1, :]
// Output layout: out[b, 0:512] = row1, out[b, 512:1024] = row2 (concat on axis 1).
//
// Data path (CDNA5 async DMA, ASYNCcnt-tracked):
//   global_load_async_to_lds_b128  (memory -> LDS, no VGPR data)
//   s_wait_asynccnt 0              (LDS portion may complete out of order)
//   global_store_async_from_lds_b128 (LDS -> memory, no VGPR data)
__global__ __launch_bounds__(256)
void VariableLengthOutputSelection_gather(const float* __restrict__ lstm1,
                                          const float* __restrict__ lstm2,
                                          const int*   __restrict__ in_len,
                                          const int*   __restrict__ sup_len,
                                          float*       __restrict__ out)
{
    __shared__ __align__(16) float lds_buf[256 * 4]; // 4 KB staging, 16 B per thread

    const int b   = blockIdx.x;
    const int tid = threadIdx.x;

    // Selected time steps for this batch row (scalar loads, wave-uniform).
    const int t1 = in_len[b]  - 1;
    const int t2 = sup_len[b] - 1;

    const float* src;
    if (tid < 128) {
        src = lstm1 + ((size_t)b * T_ + (size_t)t1) * H_ + (size_t)tid * 4;
    } else {
        src = lstm2 + ((size_t)b * T_ + (size_t)t2) * H_ + (size_t)(tid - 128) * 4;
    }
    float* dst = out + (size_t)b * (2 * H_) + (size_t)tid * 4;

    // LDS byte offset for this thread's 16B slot. Generic (flat) pointers to
    // __shared__ carry the LDS offset in bits [31:0] (aperture in [63:32]).
    uint32_t lds_off = (uint32_t)(uintptr_t)(&lds_buf[tid * 4]);

    unsigned long long gsrc = (unsigned long long)(uintptr_t)src;
    unsigned long long gdst = (unsigned long long)(uintptr_t)dst;

    // memory -> LDS (async, per-lane 16B)
    asm volatile("global_load_async_to_lds_b128 %0, %1, off"
                 :
                 : "v"(lds_off), "v"(gsrc)
                 : "memory");

    // Async load/store LDS sides are unordered: drain before reading LDS back.
    asm volatile("s_wait_asynccnt 0x0" ::: "memory");

    // LDS -> memory (async, per-lane 16B), lands directly in concat layout.
    asm volatile("global_store_async_from_lds_b128 %0, %1, off"
                 :
                 : "v"(gdst), "v"(lds_off)
                 : "memory");

    // S_ENDPGM does an implicit wait-idle, but be explicit about completion.
    asm volatile("s_wait_asynccnt 0x0" ::: "memory");
}

extern "C" void kernel_launch(void* const* d_in, const int* in_sizes, int n_in,
                              void* d_out, int out_size, void* d_ws, size_t ws_size,
                              hipStream_t stream) {
    (void)in_sizes; (void)n_in; (void)out_size; (void)d_ws; (void)ws_size;

    const float* lstm1   = (const float*)d_in[0];
    const float* lstm2   = (const float*)d_in[1];
    const int*   in_len  = (const int*)d_in[2];
    const int*   sup_len = (const int*)d_in[3];
    float*       out     = (float*)d_out;

    VariableLengthOutputSelection_gather<<<B_, 256, 0, stream>>>(
        lstm1, lstm2, in_len, sup_len, out);
}